// WeightedGCN_82231443849542
// MI455X (gfx1250) — compile-verified
//
#include <hip/hip_runtime.h>
#include <hip/hip_bf16.h>

// ---------------------------------------------------------------------------
// WeightedGCN for MI455X (gfx1250, wave32, WMMA).
// GEMMs via V_WMMA_F32_16X16X4_F32 (f32 in/out, matches reference precision;
// GEMM is ~1.9 GFLOP total, nowhere near the bottleneck, so no downcast).
// Message passing: one wave per edge, float2 gather + global_atomic_add_f32
// scatter (L2-resident working set, L2 atomic units do the aggregation).
// ---------------------------------------------------------------------------

typedef __attribute__((ext_vector_type(2))) float v2f;
typedef __attribute__((ext_vector_type(8))) float v8f;

#define N_NODES 50000
#define IN_CH   128
#define HID     64
#define BN_INV  0.9999950000374997f   // 1/sqrt(1 + 1e-5)

// ---------------------------------------------------------------------------
// Elementwise helpers
// ---------------------------------------------------------------------------
__global__ void zero_f32(float* __restrict__ p, int n) {
    int i = blockIdx.x * blockDim.x + threadIdx.x;
    if (i < n) p[i] = 0.0f;
}

// deg[row[e]] += |edge_attr[e]|
__global__ void deg_kernel(const long long* __restrict__ ei,
                           const float* __restrict__ ea,
                           float* __restrict__ deg, int n_edges) {
    int e = blockIdx.x * blockDim.x + threadIdx.x;
    if (e >= n_edges) return;
    long long r = ei[e];                       // edge_index[0][e]
    atomicAdd(&deg[r], fabsf(ea[e]));
}

// dis[i] = deg>0 ? deg^-1/2 : 0
__global__ void dis_kernel(const float* __restrict__ deg,
                           float* __restrict__ dis, int n) {
    int i = blockIdx.x * blockDim.x + threadIdx.x;
    if (i >= n) return;
    float d = deg[i];
    dis[i] = (d > 0.0f) ? rsqrtf(d) : 0.0f;
}

// agg[n][f] = bias[f]  (segment_sum target pre-loaded with conv bias)
__global__ void bias_init_kernel(float* __restrict__ agg,
                                 const float* __restrict__ b, int total) {
    int i = blockIdx.x * blockDim.x + threadIdx.x;
    if (i < total) agg[i] = b[i & (HID - 1)];
}

// in-place eval-BN + ReLU over [N, HID]
__global__ void bn_relu_kernel(float* __restrict__ h,
                               const float* __restrict__ g,
                               const float* __restrict__ be, int total) {
    int i = blockIdx.x * blockDim.x + threadIdx.x;
    if (i >= total) return;
    int f = i & (HID - 1);
    float v = h[i];
    h[i] = fmaxf(fmaf(g[f], v * BN_INV, be[f]), 0.0f);
}

// ---------------------------------------------------------------------------
// WMMA f32 GEMM:  out[M,Fout] = in[M,fin] @ W[Fout,fin]^T  (+bias, +relu)
// One wave per 16x16 output tile; blockDim = (32,8) -> 8 tiles/block in M.
// A lane layout (16x4 f32):  lane l -> row m0+(l&15), K = k + 2*(l>>4) + {0,1}
// B lane layout (4x16 f32):  lane l -> col n0+(l&15), same K pattern
// C/D (16x16 f32, 8 VGPRs):  lanes 0-15 -> M = m0+v, lanes 16-31 -> M = m0+8+v
// ---------------------------------------------------------------------------
__global__ void gemm_wmma_f32(const float* __restrict__ in, int fin,
                              const float* __restrict__ W,
                              float* __restrict__ out, int ld_out,
                              const float* __restrict__ bias, int do_relu,
                              int n_rows) {
    const int tile_m = blockIdx.x * blockDim.y + threadIdx.y;  // wave-uniform
    const int m0 = tile_m * 16;
    if (m0 >= n_rows) return;                                  // whole-wave exit
    const int n0 = blockIdx.y * 16;

    const int lane = threadIdx.x;
    const int half = lane >> 4;        // 0: K={0,1}, 1: K={2,3}
    const int lq   = lane & 15;

    const float* __restrict__ arow = in + (size_t)(m0 + lq) * fin + 2 * half;
    const float* __restrict__ brow = W  + (size_t)(n0 + lq) * fin + 2 * half;

    v8f acc = {};
#pragma unroll 4
    for (int k = 0; k < fin; k += 4) {
        v2f a = *(const v2f*)(arow + k);
        v2f b = *(const v2f*)(brow + k);
        acc = __builtin_amdgcn_wmma_f32_16x16x4_f32(
            /*neg_a=*/false, a, /*neg_b=*/false, b,
            /*c_mod=*/(short)0, acc, /*reuse_a=*/false, /*reuse_b=*/false);
    }

    const int   col   = n0 + lq;
    const float badd  = bias ? bias[col] : 0.0f;
    const int   mbase = m0 + half * 8;
#pragma unroll
    for (int v = 0; v < 8; ++v) {
        float r = acc[v] + badd;
        if (do_relu) r = fmaxf(r, 0.0f);
        out[(size_t)(mbase + v) * ld_out + col] = r;
    }
}

// ---------------------------------------------------------------------------
// Edge scatter: one wave per edge.
//   w = dis[row]*dis[col]*edge_attr[e];  out[col][:] += w * h[row][:]
// 32 lanes x float2 = 64 features; two global_atomic_add_f32 per lane.
// ---------------------------------------------------------------------------
__global__ void scatter_kernel(const long long* __restrict__ ei,
                               const float* __restrict__ ea,
                               const float* __restrict__ dis,
                               const float* __restrict__ h,
                               float* __restrict__ out, int n_edges) {
    const int e    = blockIdx.x * (blockDim.x >> 5) + (threadIdx.x >> 5);
    const int lane = threadIdx.x & 31;
    if (e >= n_edges) return;
    const long long r = ei[e];
    const long long c = ei[n_edges + e];
    const float w = dis[r] * dis[c] * ea[e];
    if (w == 0.0f) return;                        // wave-uniform skip
    const float2 hv = ((const float2*)(h + (size_t)r * HID))[lane];
    float* dst = out + (size_t)c * HID + 2 * lane;
    atomicAdd(dst + 0, w * hv.x);
    atomicAdd(dst + 1, w * hv.y);
}

// ---------------------------------------------------------------------------
// Final 32 -> 2 projection, one thread per node.
// ---------------------------------------------------------------------------
__global__ void mlp2_kernel(const float* __restrict__ hc,
                            const float* __restrict__ Wc2,
                            const float* __restrict__ bc2,
                            float* __restrict__ out, int n) {
    int i = blockIdx.x * blockDim.x + threadIdx.x;
    if (i >= n) return;
    const float* row = hc + (size_t)i * 32;
    float a0 = bc2[0], a1 = bc2[1];
#pragma unroll
    for (int k = 0; k < 32; ++k) {
        float v = row[k];
        a0 = fmaf(v, Wc2[k],      a0);
        a1 = fmaf(v, Wc2[32 + k], a1);
    }
    out[(size_t)i * 2 + 0] = a0;
    out[(size_t)i * 2 + 1] = a1;
}

// ---------------------------------------------------------------------------
// Launch
// ---------------------------------------------------------------------------
extern "C" void kernel_launch(void* const* d_in, const int* in_sizes, int n_in,
                              void* d_out, int out_size, void* d_ws, size_t ws_size,
                              hipStream_t stream) {
    const float*     x  = (const float*)d_in[0];
    const long long* ei = (const long long*)d_in[1];   // int64 [2, E]
    const float*     ea = (const float*)d_in[2];
    const float* Wl[3] = {(const float*)d_in[3], (const float*)d_in[7], (const float*)d_in[11]};
    const float* bl[3] = {(const float*)d_in[4], (const float*)d_in[8], (const float*)d_in[12]};
    const float* gl[3] = {(const float*)d_in[5], (const float*)d_in[9], (const float*)d_in[13]};
    const float* el[3] = {(const float*)d_in[6], (const float*)d_in[10], (const float*)d_in[14]};
    const float* Wc1 = (const float*)d_in[15];
    const float* bc1 = (const float*)d_in[16];
    const float* Wc2 = (const float*)d_in[17];
    const float* bc2 = (const float*)d_in[18];
    float* out = (float*)d_out;

    const int E = in_sizes[2];           // 800000
    const int N = N_NODES;               // 50000

    // Workspace carve (floats), 256B-aligned chunks
    float* deg  = (float*)d_ws;
    float* dis  = deg  + 50048;
    float* bufA = dis  + 50048;                 // [N, 64] post-linear
    float* bufB = bufA + (size_t)N * HID;       // [N, 64] aggregate / activation
    float* bufC = bufB + (size_t)N * HID;       // [N, 32] MLP hidden

    const int TPB = 256;
    const int tilesM = N / 16;                        // 3125 exactly
    dim3 gblock(32, 8);                               // 8 waves = 8 M-tiles/block
    dim3 ggrid((tilesM + 7) / 8, HID / 16);           // (391, 4)

    // Degree normalization
    zero_f32<<<(N + TPB - 1) / TPB, TPB, 0, stream>>>(deg, N);
    deg_kernel<<<(E + TPB - 1) / TPB, TPB, 0, stream>>>(ei, ea, deg, E);
    dis_kernel<<<(N + TPB - 1) / TPB, TPB, 0, stream>>>(deg, dis, N);

    const float* cur = x;
    int fin = IN_CH;
    for (int l = 0; l < 3; ++l) {
        gemm_wmma_f32<<<ggrid, gblock, 0, stream>>>(cur, fin, Wl[l], bufA, HID,
                                                    nullptr, 0, N);
        bias_init_kernel<<<(N * HID + TPB - 1) / TPB, TPB, 0, stream>>>(bufB, bl[l], N * HID);
        scatter_kernel<<<(E + 7) / 8, TPB, 0, stream>>>(ei, ea, dis, bufA, bufB, E);
        bn_relu_kernel<<<(N * HID + TPB - 1) / TPB, TPB, 0, stream>>>(bufB, gl[l], el[l], N * HID);
        cur = bufB;
        fin = HID;
    }

    // MLP head: [N,64] @ Wc1^T -> relu -> [N,32]
    dim3 mgrid((tilesM + 7) / 8, (HID / 2) / 16);     // (391, 2)
    gemm_wmma_f32<<<mgrid, gblock, 0, stream>>>(bufB, HID, Wc1, bufC, HID / 2,
                                                bc1, 1, N);
    // [N,32] @ Wc2^T -> [N,2]
    mlp2_kernel<<<(N + TPB - 1) / TPB, TPB, 0, stream>>>(bufC, Wc2, bc2, out, N);
}